// InteractionBlock_54391465837341
// MI455X (gfx1250) — compile-verified
//
#include <hip/hip_runtime.h>
#include <hip/hip_bf16.h>

// ---------------------------------------------------------------------------
// MACE-style InteractionBlock for MI455X (gfx1250, wave32, WMMA f32 16x16x4).
//
// Roofline: ~30 GFLOP f32 GEMM vs ~1 GB irregular edge gather/scatter traffic
// -> memory-bound on the edge stage. All dense contractions go to the matrix
// pipe (v_wmma_f32_16x16x4_f32); the edge MLP + gather + mid-products +
// segment-sum are fused so the (E,512) edge-weight tensor never touches HBM;
// segment-sum uses no-return global_atomic_add_f32 (STOREcnt path).
//
// Round-2 changes (VALU-per-WMMA reduction, guided by the disasm):
//  * sc einsum: outer-product operand z built in LDS chunks (kills the
//    runtime k/10, k%10 chains that dominated the inner loop), A-pairs now
//    contiguous -> single ds_load_b64 per WMMA.
//  * two output column-tiles per wave: one A load feeds two WMMAs.
//  * xv staged channel-major so vector-channel A-pairs are contiguous.
//  * out_kernel stages stride-3 v accumulator into LDS per channel.
// ---------------------------------------------------------------------------

typedef __attribute__((ext_vector_type(2))) float v2f;
typedef __attribute__((ext_vector_type(8))) float v8f;

static constexpr int NN   = 10000;   // nodes
static constexpr int NE   = 160000;  // edges
static constexpr int MULC = 128;
static constexpr int NSP  = 10;
static constexpr int NRAD = 8;

static constexpr float SILU_C    = 1.679f;
static constexpr float SC_NORM   = 0.02795084971874737f;  // 1/sqrt(1280)
static constexpr float L1_NORM   = 0.08838834764831845f;  // 1/sqrt(128)
static constexpr float L2_NORM   = 0.0625f;               // 1/sqrt(256)
static constexpr float INV_SQRT8 = 0.3535533905932738f;   // 1/sqrt(8)
static constexpr float INV_SQ3   = 0.5773502691896258f;   // 1/sqrt(3)
static constexpr float INV_NEIGH = 0.25f;                 // 1/sqrt(16)

// D = A(16x4) * B(4x16) + C, f32, wave32.
__device__ __forceinline__ v8f wmma4(v2f a, v2f b, v8f c) {
  return __builtin_amdgcn_wmma_f32_16x16x4_f32(
      /*neg_a=*/false, a, /*neg_b=*/false, b,
      /*c_mod=*/(short)0, c, /*reuse_a=*/false, /*reuse_b=*/false);
}

__device__ __forceinline__ float silu_c(float z) {
  return SILU_C * z / (1.0f + __expf(-z));
}

// ---------------------------------------------------------------------------
// Kernel 0: zero the segment-sum accumulators (s: N*256, v: N*256*3).
// ---------------------------------------------------------------------------
__global__ void zero_kernel(float* __restrict__ p, long n) {
  long i = (long)blockIdx.x * blockDim.x + threadIdx.x;
  long stride = (long)gridDim.x * blockDim.x;
  for (; i < n; i += stride) p[i] = 0.0f;
}

// ---------------------------------------------------------------------------
// Kernel 1: per 16-node tile: sc_s/sc_v (K=1280 einsum-as-GEMM, chunked
// through an LDS z buffer) into d_out, and y_s/y_v (lin1, K=128) into the
// workspace with the SAME row layout as node_feats (s in [0,128),
// v[u][c] at 128+u*3+c) so the edge gather can copy whole 512-float rows.
// 4 waves/block; wave w owns output columns [16w,16w+16) and [16w+64,16w+80).
// ---------------------------------------------------------------------------
__global__ __launch_bounds__(128) void node_kernel(
    const float* __restrict__ node_feats, const float* __restrict__ node_attrs,
    const float* __restrict__ lin1_ws, const float* __restrict__ lin1_wv,
    const float* __restrict__ sc_ws, const float* __restrict__ sc_wv,
    float* __restrict__ out, float* __restrict__ y) {
  __shared__ float xs[16][MULC];        // 8 KB  (scalar features)
  __shared__ float xvc[3][16][MULC];    // 24 KB (vector features, ch-major)
  __shared__ float at[16][16];          // 1 KB  (attrs padded 10->16)
  __shared__ float zbuf[16][256];       // 16 KB (outer-product K-chunk)

  const int n0  = blockIdx.x * 16;
  const int tid = threadIdx.x;

  for (int i = tid; i < 16 * 512; i += 128) {
    int r = i >> 9, c = i & 511;
    float vv = node_feats[(size_t)(n0 + r) * 512 + c];
    if (c < MULC) xs[r][c] = vv;
    else { int q = c - MULC; xvc[q % 3][r][q / 3] = vv; }
  }
  for (int i = tid; i < 16 * 16; i += 128) {
    int r = i >> 4, c = i & 15;
    at[r][c] = (c < NSP) ? node_attrs[(size_t)(n0 + r) * NSP + c] : 0.0f;
  }
  __syncthreads();

  const int wave = tid >> 5;    // 0..3
  const int lane = tid & 31;
  const int half = lane >> 4;   // K-half for A/B operands
  const int m    = lane & 15;   // A row
  const int nl   = lane & 15;   // B/D column
  const int wA   = wave * 16;   // first column tile
  const int wB   = wA + 64;     // second column tile

  // ---- sc: 4 targets (s, v0, v1, v2), K = 1280 in 5 chunks of 256 ----
  for (int t = 0; t < 4; ++t) {
    float (*xsrc)[MULC] = (t == 0) ? xs : xvc[t - 1];
    const float* W = (t == 0) ? sc_ws : sc_wv;
    v8f accA = {}, accB = {};
    for (int chunk = 0; chunk < 5; ++chunk) {
      __syncthreads();  // zbuf from previous chunk fully consumed
      for (int i = tid; i < 16 * 256; i += 128) {
        int r = i >> 8, kk = i & 255;
        int k = chunk * 256 + kk;
        zbuf[r][kk] = xsrc[r][k / NSP] * at[r][k % NSP];
      }
      __syncthreads();
      for (int kk = 0; kk < 256; kk += 4) {
        int k0 = kk + 2 * half;
        v2f a = { zbuf[m][k0], zbuf[m][k0 + 1] };        // -> ds_load_b64
        size_t krow = (size_t)(chunk * 256 + k0) * MULC;
        v2f bA = { W[krow + wA + nl], W[krow + MULC + wA + nl] };
        v2f bB = { W[krow + wB + nl], W[krow + MULC + wB + nl] };
        accA = wmma4(a, bA, accA);
        accB = wmma4(a, bB, accB);
      }
    }
    if (t == 0) {
#pragma unroll
      for (int r = 0; r < 8; ++r) {
        size_t row = (size_t)(n0 + r + 8 * half) * 512;
        out[row + wA + nl] = accA[r] * SC_NORM;
        out[row + wB + nl] = accB[r] * SC_NORM;
      }
    } else {
      const int c = t - 1;
#pragma unroll
      for (int r = 0; r < 8; ++r) {
        size_t row = (size_t)(n0 + r + 8 * half) * 512 + MULC;
        out[row + (wA + nl) * 3 + c] = accA[r] * SC_NORM;
        out[row + (wB + nl) * 3 + c] = accB[r] * SC_NORM;
      }
    }
  }

  // ---- y_s = xs @ lin1_ws * L1 ----
  {
    v8f accA = {}, accB = {};
    for (int kk = 0; kk < MULC; kk += 4) {
      int k0 = kk + 2 * half;
      v2f a = { xs[m][k0], xs[m][k0 + 1] };
      size_t krow = (size_t)k0 * MULC;
      v2f bA = { lin1_ws[krow + wA + nl], lin1_ws[krow + MULC + wA + nl] };
      v2f bB = { lin1_ws[krow + wB + nl], lin1_ws[krow + MULC + wB + nl] };
      accA = wmma4(a, bA, accA);
      accB = wmma4(a, bB, accB);
    }
#pragma unroll
    for (int r = 0; r < 8; ++r) {
      size_t row = (size_t)(n0 + r + 8 * half) * 512;
      y[row + wA + nl] = accA[r] * L1_NORM;
      y[row + wB + nl] = accB[r] * L1_NORM;
    }
  }
  // ---- y_v: 3 channels, A from channel-major staging (contiguous pairs) ----
  for (int c = 0; c < 3; ++c) {
    v8f accA = {}, accB = {};
    for (int kk = 0; kk < MULC; kk += 4) {
      int k0 = kk + 2 * half;
      v2f a = { xvc[c][m][k0], xvc[c][m][k0 + 1] };
      size_t krow = (size_t)k0 * MULC;
      v2f bA = { lin1_wv[krow + wA + nl], lin1_wv[krow + MULC + wA + nl] };
      v2f bB = { lin1_wv[krow + wB + nl], lin1_wv[krow + MULC + wB + nl] };
      accA = wmma4(a, bA, accA);
      accB = wmma4(a, bB, accB);
    }
#pragma unroll
    for (int r = 0; r < 8; ++r) {
      size_t row = (size_t)(n0 + r + 8 * half) * 512 + MULC;
      y[row + (wA + nl) * 3 + c] = accA[r] * L1_NORM;
      y[row + (wB + nl) * 3 + c] = accB[r] * L1_NORM;
    }
  }
}

// ---------------------------------------------------------------------------
// Kernel 2: fused edge stage. One wave per 16-edge tile:
//   gather y[src] rows into LDS (float4), 3-layer SiLU MLP via WMMA,
//   mid products, scatter via global_atomic_add_f32 (pre-scaled by 1/sqrt(16)).
// The w-quadrant of a column tile is uniform per nt, so EXEC is all-ones at
// every WMMA site.
// ---------------------------------------------------------------------------
__global__ __launch_bounds__(32) void edge_kernel(
    const float* __restrict__ y, const float* __restrict__ edge_feats,
    const float* __restrict__ edge_attrs, const int* __restrict__ edge_index,
    const float* __restrict__ fc_w0, const float* __restrict__ fc_w1,
    const float* __restrict__ fc_w2, float* __restrict__ s_acc,
    float* __restrict__ v_acc) {
  __shared__ float ytile[16][512];  // 32 KB gathered y[src]
  __shared__ float ha[16][64];      // 4 KB
  __shared__ float hb[16][64];      // 4 KB
  __shared__ float ea[16][4];
  __shared__ float ef[16][NRAD];
  __shared__ int   dstn[16];

  const int e0   = blockIdx.x * 16;
  const int lane = threadIdx.x;
  const int half = lane >> 4;
  const int m    = lane & 15;
  const int nl   = lane & 15;

  // gather y rows (src = edge_index[1])
  for (int r = 0; r < 16; ++r) {
    int sr = edge_index[NE + e0 + r];
    const float4* sp = (const float4*)(y + (size_t)sr * 512);
    float4* dp = (float4*)&ytile[r][0];
    for (int i = lane; i < 128; i += 32) dp[i] = sp[i];
  }
  for (int i = lane; i < 16 * NRAD; i += 32)
    ef[i >> 3][i & 7] = edge_feats[(size_t)e0 * NRAD + i];
  for (int i = lane; i < 16 * 4; i += 32)
    ea[i >> 2][i & 3] = edge_attrs[(size_t)e0 * 4 + i];
  if (lane < 16) dstn[lane] = edge_index[e0 + lane];  // dst = edge_index[0]
  __syncthreads();

  // layer 0: (16x8)@(8x64)
  for (int nt = 0; nt < 4; ++nt) {
    v8f acc = {};
    for (int kk = 0; kk < 8; kk += 4) {
      int k0 = kk + 2 * half;
      v2f a = { ef[m][k0], ef[m][k0 + 1] };
      v2f b = { fc_w0[k0 * 64 + nt * 16 + nl],
                fc_w0[(k0 + 1) * 64 + nt * 16 + nl] };
      acc = wmma4(a, b, acc);
    }
#pragma unroll
    for (int r = 0; r < 8; ++r)
      ha[r + 8 * half][nt * 16 + nl] = silu_c(acc[r] * INV_SQRT8);
  }
  __syncthreads();

  // layer 1: (16x64)@(64x64)
  for (int nt = 0; nt < 4; ++nt) {
    v8f acc = {};
    for (int kk = 0; kk < 64; kk += 4) {
      int k0 = kk + 2 * half;
      v2f a = { ha[m][k0], ha[m][k0 + 1] };     // -> ds_load_b64
      v2f b = { fc_w1[k0 * 64 + nt * 16 + nl],
                fc_w1[(k0 + 1) * 64 + nt * 16 + nl] };
      acc = wmma4(a, b, acc);
    }
#pragma unroll
    for (int r = 0; r < 8; ++r)
      hb[r + 8 * half][nt * 16 + nl] = silu_c(acc[r] * 0.125f);
  }
  __syncthreads();

  // layer 2 (16x64)@(64x512) fused with mid products + scatter-add
  for (int nt = 0; nt < 32; ++nt) {
    const int j0 = nt * 16;
    v8f acc = {};
    for (int kk = 0; kk < 64; kk += 4) {
      int k0 = kk + 2 * half;
      v2f a = { hb[m][k0], hb[m][k0 + 1] };
      v2f b = { fc_w2[(size_t)k0 * 512 + j0 + nl],
                fc_w2[(size_t)(k0 + 1) * 512 + j0 + nl] };
      acc = wmma4(a, b, acc);
    }
    const int j    = j0 + nl;    // column in [0,512)
    const int quad = j >> 7;     // 0=ss 1=sv 2=vs 3=vv (uniform per nt)
    const int u    = j & 127;
#pragma unroll
    for (int r = 0; r < 8; ++r) {
      const int e = r + 8 * half;
      const float wv = acc[r] * 0.125f;  // fc output /8
      const int d = dstn[e];
      const float as = ea[e][0];
      if (quad == 0) {  // mid_s0 -> s[:, u]
        atomicAdd(&s_acc[(size_t)d * 256 + u],
                  wv * ytile[e][u] * as * INV_NEIGH);
      } else if (quad == 1) {  // mid_v2 -> v[:, u, c]
        float base = wv * ytile[e][u] * INV_NEIGH;
        atomicAdd(&v_acc[((size_t)d * 256 + u) * 3 + 0], base * ea[e][1]);
        atomicAdd(&v_acc[((size_t)d * 256 + u) * 3 + 1], base * ea[e][2]);
        atomicAdd(&v_acc[((size_t)d * 256 + u) * 3 + 2], base * ea[e][3]);
      } else if (quad == 2) {  // mid_v3 -> v[:, 128+u, c]
        float base = wv * as * INV_NEIGH;
        atomicAdd(&v_acc[((size_t)d * 256 + 128 + u) * 3 + 0],
                  base * ytile[e][128 + u * 3 + 0]);
        atomicAdd(&v_acc[((size_t)d * 256 + 128 + u) * 3 + 1],
                  base * ytile[e][128 + u * 3 + 1]);
        atomicAdd(&v_acc[((size_t)d * 256 + 128 + u) * 3 + 2],
                  base * ytile[e][128 + u * 3 + 2]);
      } else {  // quad == 3: mid_s1 -> s[:, 128+u]
        float dotv = ytile[e][128 + u * 3 + 0] * ea[e][1] +
                     ytile[e][128 + u * 3 + 1] * ea[e][2] +
                     ytile[e][128 + u * 3 + 2] * ea[e][3];
        atomicAdd(&s_acc[(size_t)d * 256 + 128 + u],
                  wv * dotv * INV_SQ3 * INV_NEIGH);
      }
    }
  }
}

// ---------------------------------------------------------------------------
// Kernel 3: out += lin2 GEMMs. s staged in LDS; the stride-3 v accumulator is
// re-staged into a dense LDS tile per channel so A-pairs merge to b64.
// 4 waves/block; wave owns two column tiles.
// ---------------------------------------------------------------------------
__global__ __launch_bounds__(128) void out_kernel(
    const float* __restrict__ s_acc, const float* __restrict__ v_acc,
    const float* __restrict__ lin2_ws, const float* __restrict__ lin2_wv,
    float* __restrict__ out) {
  __shared__ float st[16][256];  // 16 KB
  __shared__ float vt[16][256];  // 16 KB (re-staged per channel)
  const int n0  = blockIdx.x * 16;
  const int tid = threadIdx.x;
  for (int i = tid; i < 16 * 256; i += 128)
    st[i >> 8][i & 255] = s_acc[(size_t)n0 * 256 + i];
  __syncthreads();

  const int wave = tid >> 5;
  const int lane = tid & 31;
  const int half = lane >> 4;
  const int m    = lane & 15;
  const int nl   = lane & 15;
  const int wA   = wave * 16;
  const int wB   = wA + 64;

  {
    v8f accA = {}, accB = {};
    for (int kk = 0; kk < 256; kk += 4) {
      int k0 = kk + 2 * half;
      v2f a = { st[m][k0], st[m][k0 + 1] };
      size_t krow = (size_t)k0 * MULC;
      v2f bA = { lin2_ws[krow + wA + nl], lin2_ws[krow + MULC + wA + nl] };
      v2f bB = { lin2_ws[krow + wB + nl], lin2_ws[krow + MULC + wB + nl] };
      accA = wmma4(a, bA, accA);
      accB = wmma4(a, bB, accB);
    }
#pragma unroll
    for (int r = 0; r < 8; ++r) {
      size_t row = (size_t)(n0 + r + 8 * half) * 512;
      out[row + wA + nl] += accA[r] * L2_NORM;
      out[row + wB + nl] += accB[r] * L2_NORM;
    }
  }
  for (int c = 0; c < 3; ++c) {
    __syncthreads();
    for (int i = tid; i < 16 * 256; i += 128) {
      int r = i >> 8, k = i & 255;
      vt[r][k] = v_acc[(((size_t)(n0 + r)) * 256 + k) * 3 + c];
    }
    __syncthreads();
    v8f accA = {}, accB = {};
    for (int kk = 0; kk < 256; kk += 4) {
      int k0 = kk + 2 * half;
      v2f a = { vt[m][k0], vt[m][k0 + 1] };
      size_t krow = (size_t)k0 * MULC;
      v2f bA = { lin2_wv[krow + wA + nl], lin2_wv[krow + MULC + wA + nl] };
      v2f bB = { lin2_wv[krow + wB + nl], lin2_wv[krow + MULC + wB + nl] };
      accA = wmma4(a, bA, accA);
      accB = wmma4(a, bB, accB);
    }
#pragma unroll
    for (int r = 0; r < 8; ++r) {
      size_t row = (size_t)(n0 + r + 8 * half) * 512 + MULC;
      out[row + (wA + nl) * 3 + c] += accA[r] * L2_NORM;
      out[row + (wB + nl) * 3 + c] += accB[r] * L2_NORM;
    }
  }
}

// ---------------------------------------------------------------------------
extern "C" void kernel_launch(void* const* d_in, const int* in_sizes, int n_in,
                              void* d_out, int out_size, void* d_ws,
                              size_t ws_size, hipStream_t stream) {
  const float* node_feats = (const float*)d_in[0];
  const float* node_attrs = (const float*)d_in[1];
  const float* edge_feats = (const float*)d_in[2];
  const float* edge_attrs = (const float*)d_in[3];
  const int*   edge_index = (const int*)d_in[4];
  const float* lin1_ws    = (const float*)d_in[5];
  const float* lin1_wv    = (const float*)d_in[6];
  const float* fc_w0      = (const float*)d_in[7];
  const float* fc_w1      = (const float*)d_in[8];
  const float* fc_w2      = (const float*)d_in[9];
  const float* lin2_ws    = (const float*)d_in[10];
  const float* lin2_wv    = (const float*)d_in[11];
  const float* sc_ws      = (const float*)d_in[12];
  const float* sc_wv      = (const float*)d_in[13];

  float* out   = (float*)d_out;
  float* y     = (float*)d_ws;                       // N*512
  float* s_acc = y + (size_t)NN * 512;               // N*256
  float* v_acc = s_acc + (size_t)NN * 256;           // N*256*3

  zero_kernel<<<2048, 256, 0, stream>>>(s_acc, (long)NN * 1024);
  node_kernel<<<NN / 16, 128, 0, stream>>>(node_feats, node_attrs, lin1_ws,
                                           lin1_wv, sc_ws, sc_wv, out, y);
  edge_kernel<<<NE / 16, 32, 0, stream>>>(y, edge_feats, edge_attrs,
                                          edge_index, fc_w0, fc_w1, fc_w2,
                                          s_acc, v_acc);
  out_kernel<<<NN / 16, 128, 0, stream>>>(s_acc, v_acc, lin2_ws, lin2_wv, out);
}